// TempAttentionBiFusion_21131239096933
// MI455X (gfx1250) — compile-verified
//
#include <hip/hip_runtime.h>
#include <math.h>

typedef float v2f __attribute__((ext_vector_type(2)));
typedef float v8f __attribute__((ext_vector_type(8)));
typedef unsigned int u32x4 __attribute__((ext_vector_type(4)));
typedef int i32x4 __attribute__((ext_vector_type(4)));
typedef int i32x8 __attribute__((ext_vector_type(8)));

#define S      8
#define C      256
#define R      64
#define BNECK  512
#define NOUT   512
#define NB     64
#define HW     196    // 14*14
#define NPIX1  169    // 13*13
#define K1     1024   // 256 in-ch * 4 taps
#define K2     2048   // 512 in-ch * 4 taps

// fp32 WMMA: D(16x16) = A(16x4) * B(4x16) + C   -> v_wmma_f32_16x16x4_f32
static __device__ __forceinline__ v8f wmma_f32(v2f a, v2f b, v8f c) {
  return __builtin_amdgcn_wmma_f32_16x16x4_f32(false, a, false, b, (short)0, c,
                                               false, false);
}

// ---------------------------------------------------------------------------
// Tensor Data Mover: flat 1D copy of 6272 f32 (= 32 channels x 196 pixels)
// from global memory into LDS.  D# per CDNA5 ISA ch.8:
//   group0: count=1 | lds_addr | global_addr[56:0] | type=2
//   group1: data_size=4B, tensor_dim0=tile_dim0=6272, tensor_dim1=tile_dim1=1,
//           tensor_dim0_stride=6272, no padding / iterate / multicast.
// Tracked with TENSORcnt; issuing wave waits, barriers publish to the block.
// ---------------------------------------------------------------------------
#if __has_include(<hip/amd_detail/amd_gfx1250_TDM.h>)
#define TDM_6ARG 1
#else
#define TDM_6ARG 0
#endif

static __device__ __forceinline__
void tdm_load_chunk(const float* src, unsigned lds_off) {
  unsigned long long ga = (unsigned long long)(uintptr_t)src;
  u32x4 g0;
  g0.x = 1u;                                   // count = 1 descriptor
  g0.y = lds_off;                              // LDS byte address
  g0.z = (unsigned)ga;                         // global_addr[31:0]
  g0.w = ((unsigned)(ga >> 32) & 0x01FFFFFFu)  // global_addr[56:32]
         | 0x80000000u;                        // type = 2 ("image")
  // group1 bitfields (see analysis): TD0=6272, TD1=1, TILE0=6272, TILE1=1
  i32x8 g1 = { 0x00020000, 0x18800000, 0x00010000, 0x18800000,
               0x00000001, 6272,       0,          0 };
  i32x4 z4 = { 0, 0, 0, 0 };
#if TDM_6ARG
  i32x8 z8 = { 0, 0, 0, 0, 0, 0, 0, 0 };
  __builtin_amdgcn_tensor_load_to_lds(g0, g1, z4, z4, z8, 0);
#else
  __builtin_amdgcn_tensor_load_to_lds(g0, g1, z4, z4, 0);
#endif
}

// ---------------------------------------------------------------------------
// Kernel 1: conv1 (2x2 VALID, 256->512ch) + bias + ReLU, fused with the four
// 12x12 shifted window-sums of the 13x13 output map.  Implicit GEMM:
// M=512 (tiled 64/block), N=169 (pad 192), K=1024.  B-matrix fed from LDS,
// staged by the TDM with double buffering.  Output s1[b][oc][tap].
// ---------------------------------------------------------------------------
__global__ __launch_bounds__(256)
void k_conv1_s1(const float* __restrict__ x, const float* __restrict__ w1,
                const float* __restrict__ b1, float* __restrict__ s1)
{
  __shared__ float lbuf[2][6400];    // 2 x (32ch x 196px) + pad
  __shared__ float lds_s1[256];      // 64 oc x 4 taps
  const int t    = threadIdx.x;
  const int lane = t & 31, wave = t >> 5;
  const int tlo  = lane & 15, thi = lane >> 4;
  const int m16  = wave & 3, nh = wave >> 2;      // wave -> (m-subtile, n-half)
  const int mtile = blockIdx.x;                   // 0..7
  const int bb    = blockIdx.y;                   // 0..63
  const int m0    = mtile * 64 + m16 * 16;
  const int nstart = nh * 6;                      // subtiles 0..5 / 6..11(pad)

  lds_s1[t] = 0.f;

  // per-j pixel geometry (p = output pixel id, oh/ow in 13x13 grid)
  int xoff[6]; float wf0[6], wf1[6], wf2[6], wf3[6];
#pragma unroll
  for (int j = 0; j < 6; ++j) {
    int p  = (nstart + j) * 16 + tlo;
    int oh = p / 13, ow = p - oh * 13;
    bool valid = (p < NPIX1);
    xoff[j] = (oh + thi) * 14 + ow;   // tap row selected by lane[4]
    wf0[j] = (valid && oh <= 11 && ow <= 11) ? 1.f : 0.f;
    wf1[j] = (valid && oh <= 11 && ow >= 1)  ? 1.f : 0.f;
    wf2[j] = (valid && oh >= 1  && ow <= 11) ? 1.f : 0.f;
    wf3[j] = (valid && oh >= 1  && ow >= 1)  ? 1.f : 0.f;
  }

  const float* xb = x + (size_t)bb * (S * C * HW);
  const float* aw = w1 + (size_t)(m0 + tlo) * K1 + 2 * thi;

  v8f acc[6];
#pragma unroll
  for (int j = 0; j < 6; ++j) acc[j] = {};

  // prologue: DMA chunk 0 into buffer 0
  if (wave == 0) {
    tdm_load_chunk(xb, (unsigned)(uintptr_t)&lbuf[0][0]);
    __builtin_amdgcn_s_wait_tensorcnt(0);
  }
  __syncthreads();

  for (int s8 = 0; s8 < 8; ++s8) {               // 8 chunks of 32 channels
    const int buf = s8 & 1;
    if (wave == 0 && s8 < 7)                     // prefetch next chunk (async)
      tdm_load_chunk(xb + (size_t)(s8 + 1) * C * HW,
                     (unsigned)(uintptr_t)&lbuf[buf ^ 1][0]);
    const float* lb = &lbuf[buf][0];
#pragma unroll 4
    for (int kc = 0; kc < 32; ++kc) {
      int ks = s8 * 32 + kc;
      v2f a = *(const v2f*)(aw + (size_t)ks * 4);  // taps (2*thi, 2*thi+1)
      const float* cb = lb + kc * HW;
#pragma unroll
      for (int j = 0; j < 6; ++j) {
        v2f bf;                                  // taps (thi,0),(thi,1)
        bf.x = cb[xoff[j]];
        bf.y = cb[xoff[j] + 1];
        acc[j] = wmma_f32(a, bf, acc[j]);
      }
    }
    __syncthreads();                             // all readers done with buf
    if (wave == 0) __builtin_amdgcn_s_wait_tensorcnt(0);
    __syncthreads();                             // publish next buffer
  }

  float bv[8];
#pragma unroll
  for (int v = 0; v < 8; ++v) bv[v] = b1[m0 + v + 8 * thi];

  float part[8][4];
#pragma unroll
  for (int v = 0; v < 8; ++v)
    part[v][0] = part[v][1] = part[v][2] = part[v][3] = 0.f;

#pragma unroll
  for (int j = 0; j < 6; ++j) {
#pragma unroll
    for (int v = 0; v < 8; ++v) {
      float r = fmaxf(acc[j][v] + bv[v], 0.f);   // bias + ReLU
      part[v][0] += wf0[j] * r;
      part[v][1] += wf1[j] * r;
      part[v][2] += wf2[j] * r;
      part[v][3] += wf3[j] * r;
    }
  }
#pragma unroll
  for (int v = 0; v < 8; ++v) {
    int base = (m16 * 16 + v + 8 * thi) * 4;
    atomicAdd(&lds_s1[base + 0], part[v][0]);
    atomicAdd(&lds_s1[base + 1], part[v][1]);
    atomicAdd(&lds_s1[base + 2], part[v][2]);
    atomicAdd(&lds_s1[base + 3], part[v][3]);
  }
  __syncthreads();
  {
    int ocl = t >> 2, wk = t & 3;
    s1[(size_t)bb * 2048 + (size_t)(mtile * 64 + ocl) * 4 + wk] = lds_s1[t];
  }
}

// ---------------------------------------------------------------------------
// Kernel 2: fused conv2 + spatial mean as GEMM  z = (w2_flat @ s1^T)/144 + b2
// M=512 (tiled 64/block), N=64 (batches), K=2048.
// ---------------------------------------------------------------------------
__global__ __launch_bounds__(256)
void k_z(const float* __restrict__ s1, const float* __restrict__ w2,
         const float* __restrict__ b2, float* __restrict__ z)
{
  const int t = threadIdx.x, lane = t & 31, wave = t >> 5;
  const int tlo = lane & 15, thi = lane >> 4;
  const int mtile = blockIdx.x;                  // 0..7
#pragma unroll
  for (int task = 0; task < 2; ++task) {
    const int tk   = wave + task * 8;            // 0..15
    const int m16  = tk & 3, nsub = tk >> 2;
    const int m0   = mtile * 64 + m16 * 16;
    const int n0   = nsub * 16;
    const float* ap = w2 + (size_t)(m0 + tlo) * K2 + 2 * thi;
    const float* bp = s1 + (size_t)(n0 + tlo) * K2 + 2 * thi;
    v8f acc = {};
#pragma unroll 8
    for (int ks = 0; ks < 512; ++ks) {
      v2f a = *(const v2f*)(ap + (size_t)ks * 4);
      v2f b = *(const v2f*)(bp + (size_t)ks * 4);
      acc = wmma_f32(a, b, acc);
    }
#pragma unroll
    for (int v = 0; v < 8; ++v) {
      int oc = m0 + v + 8 * thi;
      z[(size_t)(n0 + tlo) * BNECK + oc] = acc[v] * (1.f / 144.f) + b2[oc];
    }
  }
}

// ---------------------------------------------------------------------------
// Kernel 3: sigmoid -> temp_att, argmax over segments (first-max semantics).
// ---------------------------------------------------------------------------
__global__ void k_att(const float* __restrict__ z, float* __restrict__ att,
                      int* __restrict__ midx)
{
  int g = blockIdx.x * blockDim.x + threadIdx.x;
  if (g >= NB * R) return;
  int bb = g >> 6, r = g & 63;
  float best = -1.f; int bi = 0;
#pragma unroll
  for (int s = 0; s < S; ++s) {
    float zv = z[(size_t)bb * BNECK + s * R + r];
    float a  = 1.f / (1.f + __expf(-zv));
    att[(size_t)bb * BNECK + s * R + r] = a;
    if (a > best) { best = a; bi = s; }
  }
  midx[g] = bi;
}

// ---------------------------------------------------------------------------
// Kernel 4: per-batch rank projection (8 GEMMs of 64x196x256), fused with
// attention scaling, pivot gather, masked product, segment-sum and spatial
// mean.  total/pivot kept in VGPRs; nothing intermediate hits memory.
// ---------------------------------------------------------------------------
__global__ __launch_bounds__(256)
void k_rank(const float* __restrict__ x, const float* __restrict__ rw,
            const float* __restrict__ rb, const float* __restrict__ att,
            const int* __restrict__ midx, float* __restrict__ yb)
{
  __shared__ float lds_y[64];
  const int t = threadIdx.x, lane = t & 31, wave = t >> 5;
  const int tlo = lane & 15, thi = lane >> 4;
  const int m16 = wave & 3, nh = wave >> 2;
  const int bb = blockIdx.x;
  const int m0 = m16 * 16;
  if (t < 64) lds_y[t] = 0.f;
  __syncthreads();

  int row[8], mi[8];
#pragma unroll
  for (int v = 0; v < 8; ++v) {
    row[v] = m0 + v + 8 * thi;
    mi[v]  = midx[bb * 64 + row[v]];
  }

  float ysum[8];
#pragma unroll
  for (int v = 0; v < 8; ++v) ysum[v] = 0.f;

  for (int j = 0; j < 7; ++j) {                  // 14 padded n-subtiles / 2
    int col   = (nh * 7 + j) * 16 + tlo;
    int colc  = col < HW ? col : (HW - 1);       // clamp keeps reads in-bounds
    float cvl = (col < HW) ? 1.f : 0.f;
    v8f total = {}, pivot = {};
#pragma unroll 1
    for (int s = 0; s < S; ++s) {
      const float* xs = x + (size_t)(bb * S + s) * C * HW;
      const float* ap = rw + (size_t)s * R * C + (size_t)(m0 + tlo) * C + 2 * thi;
      __builtin_prefetch(xs + colc, 0, 0);
      v8f acc = {};
#pragma unroll 8
      for (int ks = 0; ks < 64; ++ks) {
        v2f a = *(const v2f*)(ap + ks * 4);
        const float* bx = xs + (size_t)(ks * 4 + 2 * thi) * HW + colc;
        v2f bf;
        bf.x = bx[0];
        bf.y = bx[HW];
        acc = wmma_f32(a, bf, acc);
      }
#pragma unroll
      for (int v = 0; v < 8; ++v) {
        float val = (acc[v] + rb[s * R + row[v]]) *
                    att[(size_t)bb * BNECK + s * R + row[v]];
        total[v] += val;
        pivot[v] = (s == mi[v]) ? val : pivot[v];
      }
    }
#pragma unroll
    for (int v = 0; v < 8; ++v)
      ysum[v] += cvl * pivot[v] * (total[v] - pivot[v]) * (1.f / 7.f);
  }
#pragma unroll
  for (int v = 0; v < 8; ++v) atomicAdd(&lds_y[row[v]], ysum[v]);
  __syncthreads();
  if (t < 64) yb[bb * 64 + t] = lds_y[t] * (1.f / 196.f);
}

// ---------------------------------------------------------------------------
// Kernel 5: final linear  out = yb @ lin_w^T + lin_b   (64 x 512, K=64)
// ---------------------------------------------------------------------------
__global__ void k_out(const float* __restrict__ yb, const float* __restrict__ lw,
                      const float* __restrict__ lb, float* __restrict__ out)
{
  int g = blockIdx.x * blockDim.x + threadIdx.x;
  if (g >= NB * NOUT) return;
  int bb = g >> 9, o = g & 511;
  float acc = lb[o];
  const float* yv = yb + bb * 64;
  const float* wv = lw + o * 64;
#pragma unroll 8
  for (int r = 0; r < 64; ++r) acc = fmaf(yv[r], wv[r], acc);
  out[g] = acc;
}

// ---------------------------------------------------------------------------
extern "C" void kernel_launch(void* const* d_in, const int* in_sizes, int n_in,
                              void* d_out, int out_size, void* d_ws, size_t ws_size,
                              hipStream_t stream)
{
  const float* x  = (const float*)d_in[0];   // (512,256,14,14)
  const float* w1 = (const float*)d_in[1];   // (512,256,2,2)
  const float* b1 = (const float*)d_in[2];   // (512)
  const float* w2 = (const float*)d_in[3];   // (512,512,2,2)
  const float* b2 = (const float*)d_in[4];   // (512)
  const float* rw = (const float*)d_in[5];   // (8,64,256)
  const float* rb = (const float*)d_in[6];   // (8,64)
  const float* lw = (const float*)d_in[7];   // (512,64)
  const float* lb = (const float*)d_in[8];   // (512)
  float* out = (float*)d_out;                // (64,512)

  char* ws  = (char*)d_ws;
  float* s1  = (float*)(ws);                                // 64*512*4 f32
  float* z   = (float*)(ws + 524288);                       // 64*512 f32
  float* att = (float*)(ws + 524288 + 131072);              // 64*8*64 f32
  int*   mi  = (int*)  (ws + 524288 + 262144);              // 64*64 i32
  float* yb  = (float*)(ws + 524288 + 262144 + 16384);      // 64*64 f32

  k_conv1_s1<<<dim3(8, 64), 256, 0, stream>>>(x, w1, b1, s1);
  k_z      <<<8,   256, 0, stream>>>(s1, w2, b2, z);
  k_att    <<<16,  256, 0, stream>>>(z, att, mi);
  k_rank   <<<64,  256, 0, stream>>>(x, rw, rb, att, mi, yb);
  k_out    <<<128, 256, 0, stream>>>(yb, lw, lb, out);
}